// RGATABSA_83734682403107
// MI455X (gfx1250) — compile-verified
//
#include <hip/hip_runtime.h>
#include <hip/hip_bf16.h>

// ---------------------------------------------------------------------------
// Problem constants (match reference)
// ---------------------------------------------------------------------------
#define Bb  8
#define Ss  512
#define Dd  768
#define Hh  12
#define DKk 64
#define NEGV (-1000000000.0f)

typedef __attribute__((ext_vector_type(2))) float v2f;
typedef __attribute__((ext_vector_type(8))) float v8f;
typedef __attribute__((ext_vector_type(4))) unsigned int v4u;
typedef __attribute__((ext_vector_type(8))) int v8i;
typedef __attribute__((ext_vector_type(4))) int v4i;

#if __has_builtin(__builtin_amdgcn_tensor_load_to_lds) && \
    __has_builtin(__builtin_amdgcn_s_wait_tensorcnt)
#define HAVE_TDM 1
#else
#define HAVE_TDM 0
#endif

// ---------------------------------------------------------------------------
// Workspace layout (float offsets). Q/K region reused as GCN temporaries
// after the attention stage.
// ---------------------------------------------------------------------------
static constexpr size_t O_Q      = 0;                       // 4096*768
static constexpr size_t O_K      = O_Q  + (size_t)4096*768;
static constexpr size_t O_QA     = O_K  + (size_t)4096*768;
static constexpr size_t O_KA     = O_QA + (size_t)4096*768;
static constexpr size_t O_LAMQ   = O_KA + (size_t)4096*768;         // B*H*S
static constexpr size_t O_LAMK   = O_LAMQ + (size_t)Bb*Hh*Ss;
static constexpr size_t O_ATTN   = O_LAMK + (size_t)Bb*Hh*Ss;       // B*S*S
static constexpr size_t O_ATTADJ = O_ATTN   + (size_t)Bb*Ss*Ss;
static constexpr size_t O_INTER  = O_ATTADJ + (size_t)Bb*Ss*Ss;
static constexpr size_t O_KTH    = O_INTER  + (size_t)Bb*Ss*Ss;     // B
static constexpr size_t O_INVD   = O_KTH + 16;                      // B*S
static constexpr size_t O_T      = O_Q;   // reuse (after attention)
static constexpr size_t O_H1     = O_K;   // reuse (after attention)

// ---------------------------------------------------------------------------
// LDS helpers
// ---------------------------------------------------------------------------
__device__ inline unsigned lds_offset_of(void* p) {
  return (unsigned)(unsigned long long)(__attribute__((address_space(3))) char*)p;
}

// ---------------------------------------------------------------------------
// TDM: DMA a 16 x 128 float slab (rows strided by ldb elements) into LDS.
// D# built per CDNA5 ISA ch.8:
//  group0: [1:0]=count=1, [63:32]=lds_addr, [120:64]=global_addr, [127:126]=2
//  group1: data_size=2(4B), tensor_dim0=128@bits79:48, tensor_dim1=16@111:80,
//          tile_dim0=128@127:112, tile_dim1=16@143:128, dim0_stride=ldb@207:160
//  groups 2/3 (and extra word group): zero (2-D tile)
// ---------------------------------------------------------------------------
#if HAVE_TDM
__device__ inline void tdm_load_slab(const float* gsrc, int ldb, unsigned lds_off)
{
  const unsigned long long ga = (unsigned long long)(const void*)gsrc;
  const unsigned g0w0 = 1u;                                       // count=1
  const unsigned g0w1 = lds_off;                                  // lds_addr
  const unsigned g0w2 = (unsigned)(ga & 0xFFFFFFFFull);           // addr lo
  const unsigned g0w3 = (unsigned)((ga >> 32) & 0x01FFFFFFull)    // addr[56:32]
                        | 0x80000000u;                            // type=2
  v4u g0 = { g0w0, g0w1, g0w2, g0w3 };

  const unsigned td0 = 128u, td1 = 16u;        // tensor dims (elements)
  const unsigned tl0 = 128u, tl1 = 16u;        // tile dims
  const unsigned long long st0 = (unsigned long long)(unsigned)ldb;
  const int g1w0 = (int)(2u << 16);                               // data_size=4B
  const int g1w1 = (int)((td0 & 0xFFFFu) << 16);                  // dim0[15:0]
  const int g1w2 = (int)(((td0 >> 16) & 0xFFFFu) | ((td1 & 0xFFFFu) << 16));
  const int g1w3 = (int)(((td1 >> 16) & 0xFFFFu) | ((tl0 & 0xFFFFu) << 16));
  const int g1w4 = (int)(tl1 & 0xFFFFu);                          // tile1; tile2=0
  const int g1w5 = (int)(st0 & 0xFFFFFFFFull);                    // stride0 lo
  const int g1w6 = (int)((st0 >> 32) & 0xFFFFull);                // stride0 hi
  const int g1w7 = 0;
  v8i g1 = { g1w0, g1w1, g1w2, g1w3, g1w4, g1w5, g1w6, g1w7 };
  v4i gz4 = { 0, 0, 0, 0 };
  v8i gz8 = { 0, 0, 0, 0, 0, 0, 0, 0 };
  __builtin_amdgcn_tensor_load_to_lds(g0, g1, gz4, gz4, gz8, 0);
}
#endif

// Stage one 16x128 B-slab into LDS buffer `slab` (TDM if available, else
// synchronous float4 copy by the whole block).
__device__ inline void stage_slab(const float* __restrict__ gsrc, int ldb,
                                  float* __restrict__ slab, int wave, int tid)
{
#if HAVE_TDM
  if (wave == 0) tdm_load_slab(gsrc, ldb, lds_offset_of(slab));
  (void)tid;
#else
#pragma unroll
  for (int j = 0; j < 2; ++j) {
    const int i = tid + 256 * j;          // 0..511 chunks of 16B
    const int k = i >> 5;
    const int c4 = (i & 31) << 2;
    *(float4*)(slab + k * 128 + c4) = *(const float4*)(gsrc + (size_t)k * ldb + c4);
  }
#endif
}

__device__ inline void stage_wait(int wave)
{
#if HAVE_TDM
  if (wave == 0) __builtin_amdgcn_s_wait_tensorcnt(0);
#else
  (void)wave;
#endif
  __syncthreads();
}

// ---------------------------------------------------------------------------
// Block-level fp32 WMMA GEMM, 64x128 block tile (8 waves: 4 in M x 2 in N),
// B staged through double-buffered LDS slabs of 16x128, A streamed from
// global (b64 per lane).  V_WMMA_F32_16X16X4_F32 lane layouts:
//   A 16x4 : lanes 0-15 hold M=lane, {K=0,1}; lanes 16-31 hold {K=2,3}
//   B 4x16 : lanes 0-15 hold N=lane, {K=0,1}; lanes 16-31 hold {K=2,3}
//   C 16x16: VGPR v, lanes 0-15 -> M=v, N=lane; lanes 16-31 -> M=v+8
// ---------------------------------------------------------------------------
__device__ inline void block_gemm_128(const float* __restrict__ Aw, int lda,
                                      const float* __restrict__ Bblk, int ldb,
                                      int K, v8f acc[4], int lane, int wave,
                                      int tid, float (*sB)[16 * 128])
{
  const int ar = lane & 15;
  const int kk = (lane >> 4) << 1;  // 0 or 2
  const int bn = lane & 15;
  const int wn = wave & 1;
  const int nslab = K >> 4;

  stage_slab(Bblk, ldb, sB[0], wave, tid);
  stage_wait(wave);

  for (int ks = 0; ks < nslab; ++ks) {
    const float* slab = sB[ks & 1];
    if (ks + 1 < nslab) {
      stage_slab(Bblk + (size_t)(ks + 1) * 16 * ldb, ldb, sB[(ks + 1) & 1],
                 wave, tid);
      __builtin_prefetch(&Aw[ar * lda + (ks + 1) * 16 + kk], 0, 0);
    }
#pragma unroll
    for (int k4 = 0; k4 < 16; k4 += 4) {
      const int kg = ks * 16 + k4;
      v2f a;
      a.x = Aw[ar * lda + kg + kk];
      a.y = Aw[ar * lda + kg + kk + 1];
#pragma unroll
      for (int nt = 0; nt < 4; ++nt) {
        const int c = wn * 64 + nt * 16 + bn;
        v2f b;
        b.x = slab[(k4 + kk) * 128 + c];
        b.y = slab[(k4 + kk + 1) * 128 + c];
        acc[nt] = __builtin_amdgcn_wmma_f32_16x16x4_f32(
            false, a, false, b, (short)0, acc[nt], false, false);
      }
    }
    stage_wait(wave);
  }
}

// Wave-level GEMM with B accessed transposed (B element (k, n) = Bm[n*ldb+k]);
// used for q @ k^T attention scores (K=64, data L2/L0 resident).
template <int NT>
__device__ inline void wave_gemm_bt(const float* __restrict__ A, int lda,
                                    const float* __restrict__ Bm, int ldb,
                                    int K, v8f acc[NT], int lane)
{
  const int ar = lane & 15;
  const int kk = (lane >> 4) << 1;
  const int bn = lane & 15;
  for (int k = 0; k < K; k += 4) {
    v2f a;
    a.x = A[ar * lda + k + kk];
    a.y = A[ar * lda + k + kk + 1];
#pragma unroll
    for (int nt = 0; nt < NT; ++nt) {
      const float* br = Bm + (size_t)(nt * 16 + bn) * ldb;
      v2f b;
      b.x = br[k + kk];
      b.y = br[k + kk + 1];
      acc[nt] = __builtin_amdgcn_wmma_f32_16x16x4_f32(
          false, a, false, b, (short)0, acc[nt], false, false);
    }
  }
}

__device__ inline float sigmoidf_(float x) { return 1.0f / (1.0f + __expf(-x)); }

// ---------------------------------------------------------------------------
// 1) Projections: q,k,qa,ka = (x|aspect) @ W + b, stored head-major
//    [b][h][s][dk].  Grid: (768/128, 4096/64, 4 variants), block 256.
// ---------------------------------------------------------------------------
__global__ __launch_bounds__(256) void proj_kernel(
    const float* __restrict__ x, const float* __restrict__ aspect,
    const float* __restrict__ wq, const float* __restrict__ bq,
    const float* __restrict__ wk, const float* __restrict__ bk,
    const float* __restrict__ wqa, const float* __restrict__ bqa,
    const float* __restrict__ wka, const float* __restrict__ bka,
    float* __restrict__ ws)
{
  __shared__ float sB[2][16 * 128];
  const float* X; const float* W; const float* bias; float* out;
  switch (blockIdx.z) {
    case 0:  X = x;      W = wq;  bias = bq;  out = ws + O_Q;  break;
    case 1:  X = x;      W = wk;  bias = bk;  out = ws + O_K;  break;
    case 2:  X = aspect; W = wqa; bias = bqa; out = ws + O_QA; break;
    default: X = aspect; W = wka; bias = bka; out = ws + O_KA; break;
  }
  const int tid = threadIdx.x;
  const int wave = tid >> 5, lane = tid & 31;
  const int row0 = blockIdx.y * 64 + (wave >> 1) * 16;
  const int colB = blockIdx.x * 128;

  v8f acc[4] = {};
  block_gemm_128(X + (size_t)row0 * Dd, Dd, W + colB, Dd, Dd, acc, lane, wave,
                 tid, sB);

  const int n = lane & 15, mb = (lane >> 4) * 8;
  const int col0 = colB + (wave & 1) * 64;
#pragma unroll
  for (int nt = 0; nt < 4; ++nt) {
    const int col = col0 + nt * 16 + n;
    const int h = col >> 6, dk = col & 63;
    const float bv = bias[col];
#pragma unroll
    for (int v = 0; v < 8; ++v) {
      const int row = row0 + mb + v;
      const int b = row >> 9, s = row & 511;
      out[(((size_t)(b * Hh + h) * Ss) + s) * DKk + dk] = acc[nt][v] + bv;
    }
  }
}

// ---------------------------------------------------------------------------
// 2) lambda gates: lam_q = sigmoid(qa.wlqc + blqc + q.wlqq + blqq), etc.
// ---------------------------------------------------------------------------
__global__ void lam_kernel(const float* __restrict__ wlqc, const float* __restrict__ blqc,
                           const float* __restrict__ wlqq, const float* __restrict__ blqq,
                           const float* __restrict__ wlkc, const float* __restrict__ blkc,
                           const float* __restrict__ wlkk, const float* __restrict__ blkk,
                           float* __restrict__ ws)
{
  const int idx = blockIdx.x * blockDim.x + threadIdx.x;
  if (idx >= Bb * Hh * Ss) return;
  const float* q  = ws + O_Q  + (size_t)idx * DKk;
  const float* k  = ws + O_K  + (size_t)idx * DKk;
  const float* qa = ws + O_QA + (size_t)idx * DKk;
  const float* ka = ws + O_KA + (size_t)idx * DKk;
  float aq = blqc[0] + blqq[0];
  float ak = blkc[0] + blkk[0];
#pragma unroll 8
  for (int d = 0; d < DKk; ++d) {
    aq += qa[d] * wlqc[d] + q[d] * wlqq[d];
    ak += ka[d] * wlkc[d] + k[d] * wlkk[d];
  }
  ws[O_LAMQ + idx] = sigmoidf_(aq);
  ws[O_LAMK + idx] = sigmoidf_(ak);
}

// ---------------------------------------------------------------------------
// 3) 1/(rowsum(adj)+1): one wave per row.
// ---------------------------------------------------------------------------
__global__ void denom_kernel(const float* __restrict__ adj, float* __restrict__ ws)
{
  const int wave = threadIdx.x >> 5, lane = threadIdx.x & 31;
  const int row = blockIdx.x * 8 + wave;     // B*S = 4096 rows
  const float* a = adj + (size_t)row * Ss;
  float s = 0.0f;
  for (int j = lane; j < Ss; j += 32) s += a[j];
#pragma unroll
  for (int off = 16; off >= 1; off >>= 1) s += __shfl_xor(s, off, 32);
  if (lane == 0) ws[O_INVD + row] = 1.0f / (s + 1.0f);
}

// ---------------------------------------------------------------------------
// 4) Fused attention: for a (b, 16-query tile) block, loop heads:
//    scores = q@k^T * scale (masked), softmax; score_a -> sigmoid;
//    attn += probs + lam * quasi  (lam depends on q-row only).
// ---------------------------------------------------------------------------
__global__ __launch_bounds__(256) void attn_kernel(
    const int* __restrict__ mask, float* __restrict__ ws)
{
  __shared__ float sS[16 * Ss];
  __shared__ float sA[16 * Ss];
  const int b = blockIdx.y;
  const int q0 = blockIdx.x * 16;
  const int wave = threadIdx.x >> 5, lane = threadIdx.x & 31;
  const int col0 = wave * 64;
  const float scale = 0.125f;  // 1/sqrt(64)

  float accR[2][16];
#pragma unroll
  for (int r = 0; r < 2; ++r)
#pragma unroll
    for (int j = 0; j < 16; ++j) accR[r][j] = 0.0f;

  for (int h = 0; h < Hh; ++h) {
    const size_t hb = (size_t)(b * Hh + h) * Ss;
    const float* Q  = ws + O_Q  + (hb + q0) * DKk;
    const float* Km = ws + O_K  + hb * DKk;
    const float* QA = ws + O_QA + (hb + q0) * DKk;
    const float* KA = ws + O_KA + hb * DKk;

    v8f sc[4] = {};
    v8f sa[4] = {};
    wave_gemm_bt<4>(Q,  DKk, Km + (size_t)col0 * DKk, DKk, DKk, sc, lane);
    wave_gemm_bt<4>(QA, DKk, KA + (size_t)col0 * DKk, DKk, DKk, sa, lane);

    const int n = lane & 15, mb = (lane >> 4) * 8;
#pragma unroll
    for (int nt = 0; nt < 4; ++nt) {
      const int col = col0 + nt * 16 + n;
      const bool mz = (mask[b * Ss + col] == 0);
#pragma unroll
      for (int v = 0; v < 8; ++v) {
        const int row = mb + v;
        sS[row * Ss + col] = mz ? NEGV : sc[nt][v] * scale;
        sA[row * Ss + col] = mz ? NEGV : sa[nt][v] * scale;
      }
    }
    __syncthreads();

    // softmax + accumulate: wave handles rows wave*2, wave*2+1
#pragma unroll
    for (int rr = 0; rr < 2; ++rr) {
      const int row = wave * 2 + rr;
      float vals[16];
      float mx = -3.4e38f;
#pragma unroll
      for (int j = 0; j < 16; ++j) {
        vals[j] = sS[row * Ss + lane + 32 * j];
        mx = fmaxf(mx, vals[j]);
      }
#pragma unroll
      for (int off = 16; off >= 1; off >>= 1) mx = fmaxf(mx, __shfl_xor(mx, off, 32));
      float sum = 0.0f;
#pragma unroll
      for (int j = 0; j < 16; ++j) {
        vals[j] = __expf(vals[j] - mx);
        sum += vals[j];
      }
#pragma unroll
      for (int off = 16; off >= 1; off >>= 1) sum += __shfl_xor(sum, off, 32);
      const float inv = 1.0f / sum;
      const size_t li = (size_t)(b * Hh + h) * Ss + q0 + row;
      const float lam = 1.0f - (ws[O_LAMQ + li] + ws[O_LAMK + li]);
#pragma unroll
      for (int j = 0; j < 16; ++j) {
        const float quasi = sigmoidf_(sA[row * Ss + lane + 32 * j]);
        accR[rr][j] += vals[j] * inv + lam * quasi;
      }
    }
    __syncthreads();
  }

  float* attn = ws + O_ATTN + (size_t)b * Ss * Ss;
#pragma unroll
  for (int rr = 0; rr < 2; ++rr) {
    const int row = q0 + wave * 2 + rr;
#pragma unroll
    for (int j = 0; j < 16; ++j)
      attn[(size_t)row * Ss + lane + 32 * j] = accR[rr][j];
  }
}

// ---------------------------------------------------------------------------
// 5) Exact top-k threshold per batch: 4-pass radix select on
//    order-preserving uint32 keys. One block per batch.
// ---------------------------------------------------------------------------
__device__ inline unsigned f2u_(float f) {
  unsigned u = __float_as_uint(f);
  return (u & 0x80000000u) ? ~u : (u | 0x80000000u);
}
__device__ inline float u2f_(unsigned u) {
  unsigned b = (u & 0x80000000u) ? (u & 0x7FFFFFFFu) : ~u;
  return __uint_as_float(b);
}

__global__ __launch_bounds__(256) void topk_kernel(
    const int* __restrict__ topk, float* __restrict__ ws)
{
  const int b = blockIdx.x;
  const float* flat = ws + O_ATTN + (size_t)b * Ss * Ss;
  __shared__ unsigned hist[256];
  __shared__ unsigned sPrefix;
  __shared__ int sWant;
  if (threadIdx.x == 0) { sPrefix = 0u; sWant = *topk; }
  __syncthreads();

  for (int pass = 0; pass < 4; ++pass) {
    const int shift = 24 - 8 * pass;
    hist[threadIdx.x] = 0u;
    __syncthreads();
    const unsigned pref = sPrefix;
    const unsigned pmask = (pass == 0) ? 0u : (0xFFFFFFFFu << (shift + 8));
    for (int i = threadIdx.x; i < Ss * Ss; i += 256) {
      const unsigned u = f2u_(flat[i]);
      if ((u & pmask) == pref)
        atomicAdd(&hist[(u >> shift) & 255u], 1u);
    }
    __syncthreads();
    if (threadIdx.x == 0) {
      int want = sWant;
      for (int bin = 255; bin >= 0; --bin) {
        const int c = (int)hist[bin];
        if (want > c) { want -= c; }
        else { sPrefix = pref | ((unsigned)bin << shift); break; }
      }
      sWant = want;
    }
    __syncthreads();
  }
  if (threadIdx.x == 0) ws[O_KTH + b] = u2f_(sPrefix);
}

// ---------------------------------------------------------------------------
// 6) att_adj = sel * attn  (sel = thresholded + transpose, diag forced 1);
//    inter_adj = h0*att_adj + h1*adj/2.
// ---------------------------------------------------------------------------
__global__ void attadj_kernel(const float* __restrict__ adj,
                              const float* __restrict__ hw,
                              float* __restrict__ ws)
{
  const size_t idx = (size_t)blockIdx.x * blockDim.x + threadIdx.x;
  if (idx >= (size_t)Bb * Ss * Ss) return;
  const int b = (int)(idx / (Ss * Ss));
  const int r = (int)(idx % (Ss * Ss));
  const int i = r / Ss, j = r % Ss;
  const float* attn = ws + O_ATTN;
  const float kth = ws[O_KTH + b];
  const float a = attn[idx];
  float s;
  if (i == j) {
    s = 1.0f;
  } else {
    const float at = attn[(size_t)b * Ss * Ss + (size_t)j * Ss + i];
    s = ((a >= kth) ? 1.0f : 0.0f) + ((at >= kth) ? 1.0f : 0.0f);
  }
  const float aa = s * a;
  ws[O_ATTADJ + idx] = aa;
  ws[O_INTER + idx] = hw[0] * aa + hw[1] * adj[idx] * 0.5f;
}

// ---------------------------------------------------------------------------
// 7) Batched spmm: Out[b] = Amat[b](SxS) @ Hmat[b](SxD), TDM-staged B.
//    Grid (768/128, 512/64, B), block 256.
// ---------------------------------------------------------------------------
__global__ __launch_bounds__(256) void spmm_kernel(
    const float* __restrict__ Amat, const float* __restrict__ Hmat,
    float* __restrict__ Out)
{
  __shared__ float sB[2][16 * 128];
  const int b = blockIdx.z;
  const int tid = threadIdx.x;
  const int wave = tid >> 5, lane = tid & 31;
  const int row0 = blockIdx.y * 64 + (wave >> 1) * 16;
  const int colB = blockIdx.x * 128;
  const float* A  = Amat + (size_t)b * Ss * Ss + (size_t)row0 * Ss;
  const float* Bm = Hmat + (size_t)b * Ss * Dd + colB;

  v8f acc[4] = {};
  block_gemm_128(A, Ss, Bm, Dd, Ss, acc, lane, wave, tid, sB);

  float* O = Out + (size_t)b * Ss * Dd;
  const int n = lane & 15, mb = (lane >> 4) * 8;
  const int col0 = colB + (wave & 1) * 64;
#pragma unroll
  for (int nt = 0; nt < 4; ++nt)
#pragma unroll
    for (int v = 0; v < 8; ++v)
      O[(size_t)(row0 + mb + v) * Dd + col0 + nt * 16 + n] = acc[nt][v];
}

// ---------------------------------------------------------------------------
// 8) GCN linear epilogue: Out = relu((T @ W + bias) * invd[row]).
//    T is (B*S, D) flat.  Grid (768/128, 4096/64), block 256.
// ---------------------------------------------------------------------------
__global__ __launch_bounds__(256) void gcnlin_kernel(
    const float* __restrict__ T, const float* __restrict__ W,
    const float* __restrict__ bias, const float* __restrict__ invd,
    float* __restrict__ Out)
{
  __shared__ float sB[2][16 * 128];
  const int tid = threadIdx.x;
  const int wave = tid >> 5, lane = tid & 31;
  const int row0 = blockIdx.y * 64 + (wave >> 1) * 16;
  const int colB = blockIdx.x * 128;

  v8f acc[4] = {};
  block_gemm_128(T + (size_t)row0 * Dd, Dd, W + colB, Dd, Dd, acc, lane, wave,
                 tid, sB);

  const int n = lane & 15, mb = (lane >> 4) * 8;
  const int col0 = colB + (wave & 1) * 64;
#pragma unroll
  for (int nt = 0; nt < 4; ++nt) {
    const int col = col0 + nt * 16 + n;
    const float bv = bias[col];
#pragma unroll
    for (int v = 0; v < 8; ++v) {
      const int row = row0 + mb + v;
      const float val = (acc[nt][v] + bv) * invd[row];
      Out[(size_t)row * Dd + col] = fmaxf(val, 0.0f);
    }
  }
}

// ---------------------------------------------------------------------------
// Host-side orchestration
// ---------------------------------------------------------------------------
extern "C" void kernel_launch(void* const* d_in, const int* in_sizes, int n_in,
                              void* d_out, int out_size, void* d_ws, size_t ws_size,
                              hipStream_t stream) {
  (void)in_sizes; (void)n_in; (void)out_size; (void)ws_size;
  const float* x      = (const float*)d_in[0];
  const float* aspect = (const float*)d_in[1];
  const float* adj    = (const float*)d_in[2];
  const float* wq     = (const float*)d_in[3];
  const float* bq     = (const float*)d_in[4];
  const float* wk     = (const float*)d_in[5];
  const float* bk     = (const float*)d_in[6];
  const float* wqa    = (const float*)d_in[7];
  const float* bqa    = (const float*)d_in[8];
  const float* wka    = (const float*)d_in[9];
  const float* bka    = (const float*)d_in[10];
  const float* wlqc   = (const float*)d_in[11];
  const float* blqc   = (const float*)d_in[12];
  const float* wlqq   = (const float*)d_in[13];
  const float* blqq   = (const float*)d_in[14];
  const float* wlkc   = (const float*)d_in[15];
  const float* blkc   = (const float*)d_in[16];
  const float* wlkk   = (const float*)d_in[17];
  const float* blkk   = (const float*)d_in[18];
  const float* w_syn0 = (const float*)d_in[19];
  const float* b_syn0 = (const float*)d_in[20];
  const float* w_syn1 = (const float*)d_in[21];
  const float* b_syn1 = (const float*)d_in[22];
  const float* w_sem0 = (const float*)d_in[23];
  const float* b_sem0 = (const float*)d_in[24];
  const float* w_sem1 = (const float*)d_in[25];
  const float* b_sem1 = (const float*)d_in[26];
  const float* hw     = (const float*)d_in[27];
  const int*   mask   = (const int*)d_in[28];
  const int*   topk   = (const int*)d_in[29];

  float* out = (float*)d_out;
  float* ws  = (float*)d_ws;
  const size_t BSD = (size_t)Bb * Ss * Dd;

  // Stage 1: projections (q, k, qa, ka), head-major layout
  proj_kernel<<<dim3(Dd / 128, (Bb * Ss) / 64, 4), 256, 0, stream>>>(
      x, aspect, wq, bq, wk, bk, wqa, bqa, wka, bka, ws);

  // Stage 2: lambda gates
  lam_kernel<<<(Bb * Hh * Ss + 255) / 256, 256, 0, stream>>>(
      wlqc, blqc, wlqq, blqq, wlkc, blkc, wlkk, blkk, ws);

  // Stage 3: 1/(rowsum(adj)+1)
  denom_kernel<<<(Bb * Ss) / 8, 256, 0, stream>>>(adj, ws);

  // Stage 4: fused attention -> attn (B,S,S)
  attn_kernel<<<dim3(Ss / 16, Bb), 256, 0, stream>>>(mask, ws);

  // Stage 5: exact kth-largest per batch (radix select)
  topk_kernel<<<Bb, 256, 0, stream>>>(topk, ws);

  // Stage 6: att_adj + inter_adj
  attadj_kernel<<<(unsigned)(((size_t)Bb * Ss * Ss + 255) / 256), 256, 0, stream>>>(
      adj, hw, ws);

  const dim3 spmmGrid(Dd / 128, Ss / 64, Bb);
  const dim3 linGrid(Dd / 128, (Bb * Ss) / 64);
  float* T    = ws + O_T;
  float* H1   = ws + O_H1;
  const float* invd = ws + O_INVD;

  // GCN syn: adj with w_syn
  spmm_kernel  <<<spmmGrid, 256, 0, stream>>>(adj, x, T);
  gcnlin_kernel<<<linGrid, 256, 0, stream>>>(T, w_syn0, b_syn0, invd, H1);
  spmm_kernel  <<<spmmGrid, 256, 0, stream>>>(adj, H1, T);
  gcnlin_kernel<<<linGrid, 256, 0, stream>>>(T, w_syn1, b_syn1, invd, out);

  // GCN sem: att_adj with w_sem
  spmm_kernel  <<<spmmGrid, 256, 0, stream>>>(ws + O_ATTADJ, x, T);
  gcnlin_kernel<<<linGrid, 256, 0, stream>>>(T, w_sem0, b_sem0, invd, H1);
  spmm_kernel  <<<spmmGrid, 256, 0, stream>>>(ws + O_ATTADJ, H1, T);
  gcnlin_kernel<<<linGrid, 256, 0, stream>>>(T, w_sem1, b_sem1, invd, out + BSD);

  // GCN com: inter_adj with w_syn
  spmm_kernel  <<<spmmGrid, 256, 0, stream>>>(ws + O_INTER, x, T);
  gcnlin_kernel<<<linGrid, 256, 0, stream>>>(T, w_syn0, b_syn0, invd, H1);
  spmm_kernel  <<<spmmGrid, 256, 0, stream>>>(ws + O_INTER, H1, T);
  gcnlin_kernel<<<linGrid, 256, 0, stream>>>(T, w_syn1, b_syn1, invd, out + 2 * BSD);
}